// PostProcessing2_71442486002354
// MI455X (gfx1250) — compile-verified
//
#include <hip/hip_runtime.h>
#include <math.h>

// ---- problem constants (match reference) ----
#define HH   512
#define WW   1024
#define HW   (HH*WW)          // 524288
#define BB   2
#define BHW  (BB*HW)          // 1048576
#define NC   34
#define TOPKN 200
#define N1   (TOPKN+1)        // 201
#define HP   520              // H + 2*7 - 7 + 1
#define WP   1032             // W + 2*7 - 7 + 1
#define HPWP (HP*WP)          // 536640
#define CAP  65536            // candidate list capacity per batch
#define NCENT 208             // 200 centers padded to 13*16
#define NTILE 13

typedef __attribute__((ext_vector_type(2))) float v2f;
typedef __attribute__((ext_vector_type(8))) float v8f;

// ---------------- K0: zero scratch accumulators ----------------
__global__ void k_zero(float* p, int n) {
    int i = blockIdx.x * blockDim.x + threadIdx.x;
    if (i < n) p[i] = 0.f;
}

// ---------------- K1: seg_map = argmax over 34 classes (first max) ----------------
__global__ void k_seg(const float* __restrict__ logits, float* __restrict__ seg_out) {
    int i = blockIdx.x * blockDim.x + threadIdx.x;
    if (i >= BHW) return;
    int b = i / HW, p = i % HW;
    const float* base = logits + (size_t)b * NC * HW + p;
    float best = base[0]; int bc = 0;
    #pragma unroll
    for (int c = 1; c < NC; ++c) {
        float v = base[(size_t)c * HW];
        if (v > best) { best = v; bc = c; }
    }
    seg_out[i] = (float)bc;
}

// ---------------- K2: vote scatter (round-half-even like jnp.round) ----------------
__global__ void k_vote(const float* __restrict__ reg, const float* __restrict__ seg,
                       float* __restrict__ vote) {
    int i = blockIdx.x * blockDim.x + threadIdx.x;
    if (i >= BHW) return;
    float s = seg[i];
    if (!(s > 23.99f && s <= 33.0f)) return;          // 'things' classes 24..33
    int b = i / HW, p = i % HW;
    int y = p / WW, x = p % WW;
    float ccx = (float)(x + 1) - reg[(size_t)b * 2 * HW + p];
    float ccy = (float)(y + 1) - reg[(size_t)b * 2 * HW + HW + p];
    int vx = (int)rintf(ccx), vy = (int)rintf(ccy);
    if (vx < 0 || vy < 0 || vx >= WW || vy >= HH) return;
    atomicAdd(&vote[(size_t)b * HW + vy * WW + vx], 1.f);
}

// ---------------- K3: 7x7 box conv, pad 7 -> (520,1032) aggr (an output) ----------------
__global__ void k_aggr(const float* __restrict__ vote, float* __restrict__ aggr) {
    int i = blockIdx.x * blockDim.x + threadIdx.x;
    if (i >= BB * HPWP) return;
    int b = i / HPWP, r = i % HPWP;
    int oy = r / WP, ox = r % WP;
    float s = 0.f;
    #pragma unroll
    for (int dy = 0; dy < 7; ++dy) {
        int yy = oy - 7 + dy;
        if ((unsigned)yy >= (unsigned)HH) continue;
        const float* row = vote + (size_t)b * HW + yy * WW;
        #pragma unroll
        for (int dx = 0; dx < 7; ++dx) {
            int xx = ox - 7 + dx;
            if ((unsigned)xx < (unsigned)WW) s += row[xx];
        }
    }
    aggr[i] = s;
}

// ---------------- K4: 7x7 NMS + Threshold(50) fused with candidate compaction ----------
__global__ void k_nms(const float* __restrict__ aggr, float* __restrict__ cvals,
                      int* __restrict__ cidx, int* __restrict__ ccnt) {
    int i = blockIdx.x * blockDim.x + threadIdx.x;
    if (i >= BB * HPWP) return;
    int b = i / HPWP, r = i % HPWP;
    float a = aggr[i];
    if (!(a > 50.f)) return;
    int oy = r / WP, ox = r % WP;
    float m = a;
    #pragma unroll
    for (int dy = -3; dy <= 3; ++dy) {
        int yy = oy + dy;
        if ((unsigned)yy >= (unsigned)HP) continue;
        const float* row = aggr + (size_t)b * HPWP + yy * WP;
        #pragma unroll
        for (int dx = -3; dx <= 3; ++dx) {
            int xx = ox + dx;
            if ((unsigned)xx < (unsigned)WP) m = fmaxf(m, row[xx]);
        }
    }
    if (m > a) return;                                  // keep only local maxima
    int pos = atomicAdd(&ccnt[b], 1);
    if (pos < CAP) { cvals[b * CAP + pos] = a; cidx[b * CAP + pos] = r; }
}

// ---------------- K5: top-200 by value desc, flat-index asc (stable top_k) -----------
__global__ void k_topk(float* __restrict__ cvals, const int* __restrict__ cidx,
                       const int* __restrict__ ccnt, float* __restrict__ tvals,
                       int* __restrict__ tidx) {
    int b = blockIdx.x;
    __shared__ float sv[256];
    __shared__ int   sg[256];
    __shared__ int   ss[256];
    int n = ccnt[b]; if (n > CAP) n = CAP;
    float* cv = cvals + b * CAP;
    const int* ci = cidx + b * CAP;
    int tid = threadIdx.x;
    for (int r = 0; r < TOPKN; ++r) {
        float bv = -1.f; int bg = 0x7fffffff; int bs = -1;
        for (int i = tid; i < n; i += 256) {
            float v = cv[i];
            if (v > 0.f) {
                int g = ci[i];
                if (v > bv || (v == bv && g < bg)) { bv = v; bg = g; bs = i; }
            }
        }
        sv[tid] = bv; sg[tid] = bg; ss[tid] = bs;
        __syncthreads();
        for (int st = 128; st > 0; st >>= 1) {
            if (tid < st) {
                float ov = sv[tid + st]; int og = sg[tid + st];
                if (ov > sv[tid] || (ov == sv[tid] && og < sg[tid])) {
                    sv[tid] = ov; sg[tid] = og; ss[tid] = ss[tid + st];
                }
            }
            __syncthreads();
        }
        if (tid == 0) {
            if (ss[0] >= 0) {
                tvals[b * TOPKN + r] = sv[0];
                tidx[b * TOPKN + r]  = sg[0];
                cv[ss[0]] = -1.f;                       // remove winner
            } else {
                tvals[b * TOPKN + r] = 0.f;
                tidx[b * TOPKN + r]  = 0;
            }
        }
        __syncthreads();
    }
}

// ---------------- K6: build WMMA B-operand table [-2cx | -2cy | cx^2+cy^2 | 0] --------
// Four NCENT-sized arrays; the trailing zero array lets lanes 16-31 fetch K3=0 from
// memory so the k_inst tile loop is completely branch-free.
__global__ void k_prep(const float* __restrict__ tvals, const int* __restrict__ tidx,
                       float* __restrict__ bmat) {
    int i = blockIdx.x * blockDim.x + threadIdx.x;
    if (i >= BB * NCENT) return;
    int b = i / NCENT, j = i % NCENT;
    float bx = 0.f, by = 0.f, bc = 1.0e30f;            // invalid center => huge score
    if (j < TOPKN) {
        float v = tvals[b * TOPKN + j];
        if (v > 0.f) {
            int g = tidx[b * TOPKN + j];
            float cx = (float)(g % WP), cy = (float)(g / WP);
            bx = -2.f * cx; by = -2.f * cy; bc = cx * cx + cy * cy;
        }
    }
    float* bm = bmat + b * NCENT * 4;
    bm[j]             = bx;
    bm[NCENT + j]     = by;
    bm[2 * NCENT + j] = bc;
    bm[3 * NCENT + j] = 0.f;
}

// ---------------- K7: nearest-center argmin via V_WMMA_F32_16X16X4_F32 ----------------
// S[M][N] = ccp_x*(-2cx) + ccp_y*(-2cy) + 1*(cx^2+cy^2) = dist^2 - |p|^2
// A (16x4): lanes 0-15 -> row M=lane, K0=ccp_x, K1=ccp_y; lanes 16-31 -> K2=1, K3=0
// B (4x16): lanes 0-15 -> col N=lane, K0=-2cx, K1=-2cy; lanes 16-31 -> K2=cterm, K3=0
__global__ void __launch_bounds__(256)
k_inst(const float* __restrict__ reg, const float* __restrict__ bmat,
       const float* __restrict__ seg, float* __restrict__ inst_out) {
    int wid  = (blockIdx.x * 256 + threadIdx.x) >> 5;
    int lane = threadIdx.x & 31;
    int base = wid << 4;                                // 16 pixels per wave
    int b   = base / HW;
    int loc = base % HW;
    int y = loc / WW, x = loc % WW;

    float a0, a1;
    if (lane < 16) {
        int p = loc + lane;
        a0 = (float)(x + lane + 1) - reg[(size_t)b * 2 * HW + p];
        a1 = (float)(y + 1)        - reg[(size_t)b * 2 * HW + HW + p];
    } else { a0 = 1.f; a1 = 0.f; }
    v2f A; A[0] = a0; A[1] = a1;

    int n = lane & 15;
    // branch-free per-lane operand pointers: lanes 0-15 -> {bx, by}, lanes 16-31 -> {bc, 0}
    const float* bm = bmat + b * NCENT * 4;
    int sel = (lane < 16) ? 0 : 2 * NCENT;
    const float* p0 = bm + sel + n;
    const float* p1 = bm + sel + NCENT + n;

    float bestv[8]; int besti[8];
    #pragma unroll
    for (int r = 0; r < 8; ++r) { bestv[r] = 3.0e38f; besti[r] = 0; }

    #pragma unroll
    for (int t = 0; t < NTILE; ++t) {
        v2f Bv; Bv[0] = p0[t * 16]; Bv[1] = p1[t * 16];
        v8f C = {0.f, 0.f, 0.f, 0.f, 0.f, 0.f, 0.f, 0.f};
        v8f D = __builtin_amdgcn_wmma_f32_16x16x4_f32(
            /*neg_a=*/false, A, /*neg_b=*/false, Bv,
            /*c_mod=*/(short)0, C, /*reuse_a=*/false, /*reuse_b=*/false);
        int j = t * 16 + n;
        #pragma unroll
        for (int r = 0; r < 8; ++r) {
            float d = D[r];
            if (d < bestv[r]) { bestv[r] = d; besti[r] = j; }  // strict < keeps lowest idx
        }
    }

    // cross-lane argmin over N within each 16-lane half (ties -> lower center index)
    #pragma unroll
    for (int r = 0; r < 8; ++r) {
        float v = bestv[r]; int g = besti[r];
        #pragma unroll
        for (int m = 8; m >= 1; m >>= 1) {
            float ov = __shfl_xor(v, m, 32);
            int   og = __shfl_xor(g, m, 32);
            if (ov < v || (ov == v && og < g)) { v = ov; g = og; }
        }
        bestv[r] = v; besti[r] = g;
    }

    if (n == 0) {                                       // lanes 0 (rows 0-7) and 16 (rows 8-15)
        int rowbase = base + ((lane >> 4) << 3);
        #pragma unroll
        for (int r = 0; r < 8; ++r) {
            int p = rowbase + r;
            float s = seg[p];
            int id = (s > 23.99f && s <= 33.0f) ? (besti[r] + 1) : 0;
            inst_out[p] = (float)id;
        }
    }
}

// ---------------- K8: per-instance pixel counts + class histogram ----------------
__global__ void k_stats(const float* __restrict__ inst, const float* __restrict__ seg,
                        float* __restrict__ cnt, float* __restrict__ hist) {
    int i = blockIdx.x * blockDim.x + threadIdx.x;
    if (i >= BHW) return;
    int b  = i / HW;
    int id = (int)inst[i];
    int c  = (int)seg[i];
    atomicAdd(&cnt[b * N1 + id], 1.f);
    atomicAdd(&hist[(size_t)(b * N1 + id) * NC + c], 1.f);
}

// ---------------- K9: majority class per instance; emit counts & inst_cls -----------
__global__ void k_cls(const float* __restrict__ cnt, const float* __restrict__ hist,
                      int* __restrict__ icls, float* __restrict__ out_cnt,
                      float* __restrict__ out_icls) {
    int i = blockIdx.x * blockDim.x + threadIdx.x;
    if (i >= BB * N1) return;
    int b = i / N1, j = i % N1;
    const float* h = hist + (size_t)i * NC;
    float best = h[0]; int bc = 0;
    #pragma unroll
    for (int c = 1; c < NC; ++c) {
        float v = h[c];
        if (v > best) { best = v; bc = c; }
    }
    icls[i] = bc;
    if (j >= 1) {
        out_cnt [b * TOPKN + j - 1] = cnt[i];
        out_icls[b * TOPKN + j - 1] = (float)bc;
    }
}

// ---------------- K10: softmax prob of instance's majority class, accumulate --------
__global__ void k_prob(const float* __restrict__ logits, const float* __restrict__ inst,
                       const int* __restrict__ icls, float* __restrict__ pacc) {
    int i = blockIdx.x * blockDim.x + threadIdx.x;
    if (i >= BHW) return;
    int b = i / HW, p = i % HW;
    int id = (int)inst[i];
    int c  = icls[b * N1 + id];
    const float* base = logits + (size_t)b * NC * HW + p;
    float m = base[0];
    #pragma unroll
    for (int k = 1; k < NC; ++k) m = fmaxf(m, base[(size_t)k * HW]);
    float den = 0.f, num = 0.f;
    #pragma unroll
    for (int k = 0; k < NC; ++k) {
        float e = __expf(base[(size_t)k * HW] - m);
        den += e;
        if (k == c) num = e;
    }
    atomicAdd(&pacc[b * N1 + id], num / den);
}

// ---------------- K11: seg_prob = probsum / max(count, 1) ----------------
__global__ void k_fin(const float* __restrict__ pacc, const float* __restrict__ cnt,
                      float* __restrict__ out_prob) {
    int i = blockIdx.x * blockDim.x + threadIdx.x;
    if (i >= BB * TOPKN) return;
    int b = i / TOPKN, j = i % TOPKN;
    out_prob[i] = pacc[b * N1 + j + 1] / fmaxf(cnt[b * N1 + j + 1], 1.f);
}

extern "C" void kernel_launch(void* const* d_in, const int* in_sizes, int n_in,
                              void* d_out, int out_size, void* d_ws, size_t ws_size,
                              hipStream_t stream) {
    (void)in_sizes; (void)n_in; (void)out_size; (void)ws_size;
    const float* logits = (const float*)d_in[0];
    const float* reg    = (const float*)d_in[2];     // d_in[1] placeholder unused
    float* out = (float*)d_out;
    float* ws  = (float*)d_ws;

    // output layout (floats)
    float* o_inst    = out;                               // BHW
    float* o_seg     = out + (size_t)BHW;                 // BHW
    float* o_aggr    = out + (size_t)2 * BHW;             // BB*HPWP
    float* o_cntout  = o_aggr + (size_t)BB * HPWP;        // BB*TOPKN
    float* o_iclsout = o_cntout + (size_t)BB * TOPKN;     // BB*TOPKN
    float* o_probout = o_iclsout + (size_t)BB * TOPKN;    // BB*TOPKN

    // workspace layout (float units)
    size_t o_vote = 0;                                    // BHW (zeroed)
    size_t o_cnt  = o_vote + (size_t)BHW;                 // BB*N1 (zeroed)
    size_t o_hist = o_cnt + (size_t)BB * N1;              // BB*N1*NC (zeroed)
    size_t o_pacc = o_hist + (size_t)BB * N1 * NC;        // BB*N1 (zeroed)
    size_t o_ccnt = o_pacc + (size_t)BB * N1;             // BB ints (zeroed)
    size_t zn     = o_ccnt + (size_t)BB;                  // zero region size
    size_t o_cv   = zn;                                   // BB*CAP
    size_t o_ci   = o_cv + (size_t)BB * CAP;              // BB*CAP ints
    size_t o_tv   = o_ci + (size_t)BB * CAP;              // BB*TOPKN
    size_t o_ti   = o_tv + (size_t)BB * TOPKN;            // BB*TOPKN ints
    size_t o_bm   = o_ti + (size_t)BB * TOPKN;            // BB*NCENT*4
    size_t o_ic   = o_bm + (size_t)BB * NCENT * 4;        // BB*N1 ints

    k_zero<<<(int)((zn + 255) / 256), 256, 0, stream>>>(ws, (int)zn);
    k_seg <<<(BHW + 255) / 256, 256, 0, stream>>>(logits, o_seg);
    k_vote<<<(BHW + 255) / 256, 256, 0, stream>>>(reg, o_seg, ws + o_vote);
    k_aggr<<<(BB * HPWP + 255) / 256, 256, 0, stream>>>(ws + o_vote, o_aggr);
    k_nms <<<(BB * HPWP + 255) / 256, 256, 0, stream>>>(o_aggr, ws + o_cv,
                                                        (int*)(ws + o_ci), (int*)(ws + o_ccnt));
    k_topk<<<BB, 256, 0, stream>>>(ws + o_cv, (int*)(ws + o_ci), (int*)(ws + o_ccnt),
                                   ws + o_tv, (int*)(ws + o_ti));
    k_prep<<<(BB * NCENT + 255) / 256, 256, 0, stream>>>(ws + o_tv, (int*)(ws + o_ti), ws + o_bm);
    k_inst<<<BHW / 128, 256, 0, stream>>>(reg, ws + o_bm, o_seg, o_inst);
    k_stats<<<(BHW + 255) / 256, 256, 0, stream>>>(o_inst, o_seg, ws + o_cnt, ws + o_hist);
    k_cls <<<(BB * N1 + 255) / 256, 256, 0, stream>>>(ws + o_cnt, ws + o_hist,
                                                      (int*)(ws + o_ic), o_cntout, o_iclsout);
    k_prob<<<(BHW + 255) / 256, 256, 0, stream>>>(logits, o_inst, (int*)(ws + o_ic), ws + o_pacc);
    k_fin <<<(BB * TOPKN + 255) / 256, 256, 0, stream>>>(ws + o_pacc, ws + o_cnt, o_probout);
}